// Film_21363167331062
// MI455X (gfx1250) — compile-verified
//
#include <hip/hip_runtime.h>

typedef __attribute__((ext_vector_type(2))) float v2f;
typedef __attribute__((ext_vector_type(8))) float v8f;

#define NB    65536
#define SV    192
#define CMD   96
#define SV2   384
#define K1BLK 512    // kernel1: 128 rows per block
#define K3BLK 1024   // kernel3: 64 rows per block

__device__ __forceinline__ float relu_(float x){ return x > 0.f ? x : 0.f; }

// ---------------------------------------------------------------------------
// Kernel 1 (stats pass): cm layernorm -> h = relu(cm_ln @ film_W.T + b) via
// fp32 WMMA, accumulate deterministic per-block column sum/sumsq for BN.
// h is NOT stored (recomputed in kernel 3) -> saves ~200 MB of HBM traffic.
// ---------------------------------------------------------------------------
__global__ __launch_bounds__(256)
void film_ln_stats(const float* __restrict__ cm_in,
                   const float* __restrict__ cm_g, const float* __restrict__ cm_b,
                   const float* __restrict__ film_W, const float* __restrict__ film_b,
                   float* __restrict__ partials)
{
  __shared__ float cmn[128 * 100];   // padded stride 100 -> conflict-free ds_load_b64
  __shared__ float psum[SV2];
  __shared__ float psq[SV2];
  const int tid  = threadIdx.x;
  const int row0 = blockIdx.x * 128;

  for (int idx = tid; idx < 128 * CMD; idx += 256) {
    int r = idx / CMD, c = idx - r * CMD;
    cmn[r * 100 + c] = cm_in[(size_t)(row0 + r) * CMD + c];
  }
  __syncthreads();

  { // per-row layernorm: 2 threads per row, 48 elems each
    int r = tid >> 1, hh = (tid & 1) * 48;
    float* rp = &cmn[r * 100 + hh];
    float s = 0.f;
#pragma unroll
    for (int k = 0; k < 48; ++k) s += rp[k];
    s += __shfl_xor(s, 1, 32);
    float mu = s * (1.0f / 96.0f);
    float vv = 0.f;
#pragma unroll
    for (int k = 0; k < 48; ++k) { float d = rp[k] - mu; vv += d * d; }
    vv += __shfl_xor(vv, 1, 32);
    float rs = rsqrtf(vv * (1.0f / 96.0f) + 1e-5f);
#pragma unroll
    for (int k = 0; k < 48; ++k) {
      int c = hh + k;
      rp[k] = (rp[k] - mu) * rs * cm_g[c] + cm_b[c];
    }
  }
  __syncthreads();

  const int wave = tid >> 5, lane = tid & 31;
  const int lh = lane >> 4, li = lane & 15;

  // wave w owns col-tiles 3w..3w+2 -> exclusive columns, atomic-free partials
  for (int ct = wave * 3; ct < wave * 3 + 3; ++ct) {
    const int n = ct * 16 + li;
    v2f bf[24];                              // hoisted B fragments, whole K=96
#pragma unroll
    for (int kk = 0; kk < 24; ++kk) {
      const float* wp = &film_W[n * CMD + kk * 4 + lh * 2];
      bf[kk].x = wp[0]; bf[kk].y = wp[1];
    }
    const float bias = film_b[n];
    float colsum = 0.f, colsq = 0.f;
    for (int rt = 0; rt < 8; ++rt) {
      v8f acc = {0.f,0.f,0.f,0.f,0.f,0.f,0.f,0.f};
      const float* ap = &cmn[(rt * 16 + li) * 100 + lh * 2];
#pragma unroll
      for (int kk = 0; kk < 24; ++kk) {
        v2f af; af.x = ap[kk * 4]; af.y = ap[kk * 4 + 1];
        acc = __builtin_amdgcn_wmma_f32_16x16x4_f32(false, af, false, bf[kk],
                                                    (short)0, acc, false, false);
      }
#pragma unroll
      for (int r = 0; r < 8; ++r) {
        float v = relu_(acc[r] + bias);
        colsum += v; colsq += v * v;
      }
    }
    colsum += __shfl_xor(colsum, 16, 32);
    colsq  += __shfl_xor(colsq, 16, 32);
    if (lh == 0) { psum[n] = colsum; psq[n] = colsq; }
  }
  __syncthreads();
  for (int j = tid; j < SV2; j += 256) {
    partials[(size_t)blockIdx.x * (2 * SV2) + j]       = psum[j];
    partials[(size_t)blockIdx.x * (2 * SV2) + SV2 + j] = psq[j];
  }
}

// ---------------------------------------------------------------------------
// Kernel 2: fold partials -> per-column scale/shift (batchnorm affine).
// ---------------------------------------------------------------------------
__global__ __launch_bounds__(SV2)
void bn_reduce(const float* __restrict__ partials,
               const float* __restrict__ bn_g, const float* __restrict__ bn_b,
               float* __restrict__ scaleshift)
{
  int j = threadIdx.x;                  // 0..383
  float s = 0.f, sq = 0.f;
  for (int b = 0; b < K1BLK; ++b) {
    s  += partials[(size_t)b * (2 * SV2) + j];
    sq += partials[(size_t)b * (2 * SV2) + SV2 + j];
  }
  float mu  = s  * (1.0f / (float)NB);
  float var = sq * (1.0f / (float)NB) - mu * mu;
  float sc  = bn_g[j] * rsqrtf(var + 1e-5f);
  scaleshift[j]       = sc;
  scaleshift[SV2 + j] = bn_b[j] - mu * sc;
}

// ---------------------------------------------------------------------------
// fp32 WMMA GEMM stage: dst = act(src @ W.T + b), src/dst in LDS (stride 196),
// W row-major [192][192] from global (L2-resident, reused by all blocks).
// ---------------------------------------------------------------------------
__device__ __forceinline__
void gemm192(const float* __restrict__ src, float* __restrict__ dst,
             const float* __restrict__ W, const float* __restrict__ bvec,
             bool do_relu, int tid)
{
  const int wave = tid >> 5, lane = tid & 31;
  const int lh = lane >> 4, li = lane & 15;
  for (int tile = wave; tile < 48; tile += 8) {     // 4 row-tiles x 12 col-tiles
    const int rt = tile & 3, ct = tile >> 2;
    const int n  = ct * 16 + li;
    v8f acc = {0.f,0.f,0.f,0.f,0.f,0.f,0.f,0.f};
    const float* ap = &src[(rt * 16 + li) * 196 + lh * 2];
    const float* wp = &W[n * SV + lh * 2];
#pragma unroll 8
    for (int kk = 0; kk < 48; ++kk) {
      v2f af; af.x = ap[kk * 4]; af.y = ap[kk * 4 + 1];
      v2f bf; bf.x = wp[kk * 4]; bf.y = wp[kk * 4 + 1];
      acc = __builtin_amdgcn_wmma_f32_16x16x4_f32(false, af, false, bf,
                                                  (short)0, acc, false, false);
    }
    const float bias = bvec[n];
    const int mb = rt * 16 + lh * 8;
#pragma unroll
    for (int r = 0; r < 8; ++r) {
      float v = acc[r] + bias;
      if (do_relu) v = relu_(v);
      dst[(mb + r) * 196 + n] = v;
    }
  }
}

// ---------------------------------------------------------------------------
// Film recompute + modulation: per tile, run TWO fp32 WMMA accumulators
// (gamma columns n, beta columns n+192) sharing one A-fragment stream, apply
// BN scale/shift and r1 = relu(gamma * ln_asv + beta) -> LDS.
// ---------------------------------------------------------------------------
__device__ __forceinline__
void film_modulate(const float* __restrict__ cmn, const float* __restrict__ lnasv,
                   float* __restrict__ r1,
                   const float* __restrict__ film_W, const float* __restrict__ film_b,
                   const float* __restrict__ scaleshift, int tid)
{
  const int wave = tid >> 5, lane = tid & 31;
  const int lh = lane >> 4, li = lane & 15;
  for (int tile = wave; tile < 48; tile += 8) {     // 4 row-tiles x 12 col-tiles
    const int rt = tile & 3, ct = tile >> 2;
    const int n  = ct * 16 + li;                    // gamma column
    const int nb = n + SV;                          // beta column
    v8f accg = {0.f,0.f,0.f,0.f,0.f,0.f,0.f,0.f};
    v8f accb = {0.f,0.f,0.f,0.f,0.f,0.f,0.f,0.f};
    const float* ap = &cmn[(rt * 16 + li) * 100 + lh * 2];
    const float* wg = &film_W[n  * CMD + lh * 2];
    const float* wb = &film_W[nb * CMD + lh * 2];
#pragma unroll 6
    for (int kk = 0; kk < 24; ++kk) {
      v2f af; af.x = ap[kk * 4]; af.y = ap[kk * 4 + 1];
      v2f bg; bg.x = wg[kk * 4]; bg.y = wg[kk * 4 + 1];
      v2f bb; bb.x = wb[kk * 4]; bb.y = wb[kk * 4 + 1];
      accg = __builtin_amdgcn_wmma_f32_16x16x4_f32(false, af, false, bg,
                                                   (short)0, accg, false, false);
      accb = __builtin_amdgcn_wmma_f32_16x16x4_f32(false, af, false, bb,
                                                   (short)0, accb, false, false);
    }
    const float scg = scaleshift[n],  shg = scaleshift[SV2 + n];
    const float scb = scaleshift[nb], shb = scaleshift[SV2 + nb];
    const float bgc = film_b[n], bbc = film_b[nb];
    const int mb = rt * 16 + lh * 8;
#pragma unroll
    for (int r = 0; r < 8; ++r) {
      float gamma = relu_(accg[r] + bgc) * scg + shg;
      float beta  = relu_(accb[r] + bbc) * scb + shb;
      float ln    = lnasv[(mb + r) * 196 + n];
      r1[(mb + r) * 196 + n] = relu_(gamma * ln + beta);
    }
  }
}

// ---------------------------------------------------------------------------
// Kernel 3: LN(cm) + LN(asv) -> film recompute + FiLM -> GEMM(em1) -> relu ->
//           GEMM(em2) -> softmax gate -> mix -> cosine -> sigmoid.
// ---------------------------------------------------------------------------
__global__ __launch_bounds__(256)
void fuse_all(const float* __restrict__ asv_enr, const float* __restrict__ asv_tst,
              const float* __restrict__ cm_tst,
              const float* __restrict__ sv_g, const float* __restrict__ sv_b,
              const float* __restrict__ cm_g, const float* __restrict__ cm_b,
              const float* __restrict__ film_W, const float* __restrict__ film_b,
              const float* __restrict__ em1_W, const float* __restrict__ em1_b,
              const float* __restrict__ em2_W, const float* __restrict__ em2_b,
              const float* __restrict__ probs_W, const float* __restrict__ probs_b,
              const float* __restrict__ scaleshift,
              float* __restrict__ out)
{
  __shared__ float bufA[64 * 196];   // r1, later e2
  __shared__ float bufB[64 * 196];   // ln(asv_tst), later t
  __shared__ float cmn[64 * 100];    // ln(cm_tst), padded
  const int tid  = threadIdx.x;
  const int row0 = blockIdx.x * 64;

  for (int idx = tid; idx < 64 * SV; idx += 256) {
    int r = idx / SV, c = idx - r * SV;
    bufB[r * 196 + c] = asv_tst[(size_t)(row0 + r) * SV + c];
  }
  for (int idx = tid; idx < 64 * CMD; idx += 256) {
    int r = idx / CMD, c = idx - r * CMD;
    cmn[r * 100 + c] = cm_tst[(size_t)(row0 + r) * CMD + c];
  }
  __syncthreads();

  { // layernorm asv rows in place (4 threads per row, 48 elems each)
    int r = tid >> 2, q = (tid & 3) * 48;
    float* rp = &bufB[r * 196 + q];
    float s = 0.f;
#pragma unroll
    for (int k = 0; k < 48; ++k) s += rp[k];
    s += __shfl_xor(s, 1, 32); s += __shfl_xor(s, 2, 32);
    float mu = s * (1.0f / 192.0f);
    float vv = 0.f;
#pragma unroll
    for (int k = 0; k < 48; ++k) { float d = rp[k] - mu; vv += d * d; }
    vv += __shfl_xor(vv, 1, 32); vv += __shfl_xor(vv, 2, 32);
    float rs = rsqrtf(vv * (1.0f / 192.0f) + 1e-5f);
#pragma unroll
    for (int k = 0; k < 48; ++k) {
      int c = q + k;
      rp[k] = (rp[k] - mu) * rs * sv_g[c] + sv_b[c];
    }
  }
  { // layernorm cm rows in place (4 threads per row, 24 elems each)
    int r = tid >> 2, q = (tid & 3) * 24;
    float* rp = &cmn[r * 100 + q];
    float s = 0.f;
#pragma unroll
    for (int k = 0; k < 24; ++k) s += rp[k];
    s += __shfl_xor(s, 1, 32); s += __shfl_xor(s, 2, 32);
    float mu = s * (1.0f / 96.0f);
    float vv = 0.f;
#pragma unroll
    for (int k = 0; k < 24; ++k) { float d = rp[k] - mu; vv += d * d; }
    vv += __shfl_xor(vv, 1, 32); vv += __shfl_xor(vv, 2, 32);
    float rs = rsqrtf(vv * (1.0f / 96.0f) + 1e-5f);
#pragma unroll
    for (int k = 0; k < 24; ++k) {
      int c = q + k;
      rp[k] = (rp[k] - mu) * rs * cm_g[c] + cm_b[c];
    }
  }
  __syncthreads();

  film_modulate(cmn, bufB, bufA, film_W, film_b, scaleshift, tid);  // -> r1 in bufA
  __syncthreads();
  gemm192(bufA, bufB, em1_W, em1_b, /*relu=*/true,  tid);   // t  = relu(r1@em1.T+b1)
  __syncthreads();
  gemm192(bufB, bufA, em2_W, em2_b, /*relu=*/false, tid);   // e2 = t@em2.T+b2
  __syncthreads();

  { // softmax gate + mix + cosine + sigmoid (4 threads per row)
    int r = tid >> 2, q = tid & 3;
    size_t grow = (size_t)(row0 + r);
    const float* cmrow = &cm_tst[grow * CMD];
    float l0 = 0.f, l1 = 0.f;
    for (int k = q * 24; k < q * 24 + 24; ++k) {
      float c = cmrow[k];
      l0 += c * probs_W[k];
      l1 += c * probs_W[CMD + k];
    }
    l0 += __shfl_xor(l0, 1, 32); l0 += __shfl_xor(l0, 2, 32);
    l1 += __shfl_xor(l1, 1, 32); l1 += __shfl_xor(l1, 2, 32);
    l0 += probs_b[0]; l1 += probs_b[1];
    float m  = fmaxf(l0, l1);
    float e0 = __expf(l0 - m), e1 = __expf(l1 - m);
    float inv = 1.0f / (e0 + e1);
    float p0 = e0 * inv, p1 = e1 * inv;

    const float* trow = &asv_tst[grow * SV];      // L2 hit (staged earlier)
    const float* erow = &asv_enr[grow * SV];
    float dxe = 0.f, dxx = 0.f, dee = 0.f;
    for (int k = q * 48; k < q * 48 + 48; ++k) {
      float x = p0 * trow[k] + p1 * bufA[r * 196 + k];
      float e = erow[k];
      dxe += x * e; dxx += x * x; dee += e * e;
    }
    dxe += __shfl_xor(dxe, 1, 32); dxe += __shfl_xor(dxe, 2, 32);
    dxx += __shfl_xor(dxx, 1, 32); dxx += __shfl_xor(dxx, 2, 32);
    dee += __shfl_xor(dee, 1, 32); dee += __shfl_xor(dee, 2, 32);
    if (q == 0) {
      float nx  = fmaxf(sqrtf(dxx), 1e-8f);
      float ne  = fmaxf(sqrtf(dee), 1e-8f);
      float cos = dxe / (nx * ne);
      out[grow] = 1.0f / (1.0f + __expf(-cos));
    }
  }
}

// ---------------------------------------------------------------------------
extern "C" void kernel_launch(void* const* d_in, const int* in_sizes, int n_in,
                              void* d_out, int out_size, void* d_ws, size_t ws_size,
                              hipStream_t stream)
{
  const float* asv_enr = (const float*)d_in[0];
  const float* asv_tst = (const float*)d_in[1];
  const float* cm_tst  = (const float*)d_in[2];
  const float* sv_g    = (const float*)d_in[3];
  const float* sv_b    = (const float*)d_in[4];
  const float* cm_g    = (const float*)d_in[5];
  const float* cm_b    = (const float*)d_in[6];
  const float* film_W  = (const float*)d_in[7];
  const float* film_b  = (const float*)d_in[8];
  const float* bn_g    = (const float*)d_in[9];
  const float* bn_b    = (const float*)d_in[10];
  const float* probs_W = (const float*)d_in[11];
  const float* probs_b = (const float*)d_in[12];
  const float* em1_W   = (const float*)d_in[13];
  const float* em1_b   = (const float*)d_in[14];
  const float* em2_W   = (const float*)d_in[15];
  const float* em2_b   = (const float*)d_in[16];
  float* out = (float*)d_out;

  float* ws         = (float*)d_ws;
  float* partials   = ws;                                  // K1BLK * 2*SV2 floats
  float* scaleshift = partials + (size_t)K1BLK * 2 * SV2;  // 2*SV2 floats (~1.6 MB total)

  film_ln_stats<<<K1BLK, 256, 0, stream>>>(cm_tst, cm_g, cm_b, film_W, film_b,
                                           partials);
  bn_reduce<<<1, SV2, 0, stream>>>(partials, bn_g, bn_b, scaleshift);
  fuse_all<<<K3BLK, 256, 0, stream>>>(asv_enr, asv_tst, cm_tst, sv_g, sv_b,
                                      cm_g, cm_b, film_W, film_b,
                                      em1_W, em1_b, em2_W, em2_b,
                                      probs_W, probs_b, scaleshift, out);
}